// YOLOV5TorchObjectDetector_7224134992517
// MI455X (gfx1250) — compile-verified
//
#include <hip/hip_runtime.h>
#include <cstdint>

// ---------------------------------------------------------------------------
// YOLOv5 NMS post-processing for MI455X (gfx1250, wave32).
//   inputs : prediction (16,25200,85) f32, logits (16,25200,80) f32
//   outputs: det (16,300,6) | mask (16,300) | logits_out (16,300,80), flat f32
// ---------------------------------------------------------------------------

#define NB        16
#define NPRED     25200
#define NCLS      80
#define ROWF      85            // 4 box + 1 obj + 80 cls
#define KTOP      1000
#define MAXDET    300
#define CONF_T    0.45f
#define IOU_T     0.45f
#define TILE_ROWS 256
#define SORT_N    32768         // next pow2 >= 25200
#define NWORD     32            // 32 x 32-bit words >= KTOP bits (wave32 fit)

typedef unsigned u32x4 __attribute__((ext_vector_type(4)));
typedef int      i32x4 __attribute__((ext_vector_type(4)));
typedef int      i32x8 __attribute__((ext_vector_type(8)));

// ---------------------------------------------------------------------------
// Kernel 1: stream candidates through LDS (async global->LDS b128), compute
// score / class-argmax / xyxy box per candidate.
// rows*85 is divisible by 4 for every tile (256*85 and 112*85), so pure b128.
// ---------------------------------------------------------------------------
__global__ __launch_bounds__(TILE_ROWS) void k_prep(const float* __restrict__ pred,
                                                    float* __restrict__ score,
                                                    float* __restrict__ boxes,
                                                    int*   __restrict__ cls) {
  extern __shared__ float tile[];                     // TILE_ROWS * 85 floats
  const int img  = blockIdx.y;
  const int row0 = blockIdx.x * TILE_ROWS;
  const int rows = min(TILE_ROWS, NPRED - row0);
  if (rows <= 0) return;

  const float* gbase = pred + ((size_t)img * NPRED + row0) * ROWF;
  const int nvec = (rows * ROWF) >> 2;                // 16B chunks
  const unsigned ldsBase = (unsigned)(uintptr_t)tile; // LDS offset = addr[31:0]

  for (int v = threadIdx.x; v < nvec; v += blockDim.x) {
    unsigned lds = ldsBase + (unsigned)v * 16u;
    unsigned long long ga = (unsigned long long)(uintptr_t)(gbase + (size_t)v * 4);
    asm volatile("global_load_async_to_lds_b128 %0, %1, off"
                 :: "v"(lds), "v"(ga) : "memory");
  }
  asm volatile("s_wait_asynccnt 0" ::: "memory");     // my wave's copies landed
  __syncthreads();                                    // everyone's copies landed

  const int t = threadIdx.x;
  if (t < rows) {
    const float* r = tile + (size_t)t * ROWF;
    const float obj = r[4];
    float best = -1.0f; int bc = 0;
    #pragma unroll 8
    for (int c = 0; c < NCLS; ++c) {
      float v = r[5 + c] * obj;
      if (v > best) { best = v; bc = c; }             // strict > == first argmax
    }
    const float cx = r[0], cy = r[1], w = r[2], h = r[3];
    const bool valid = (obj > CONF_T) && (best > CONF_T);

    const size_t g = (size_t)img * NPRED + (row0 + t);
    score[g]        = valid ? best : -1.0f;
    boxes[g*4 + 0]  = cx - 0.5f * w;
    boxes[g*4 + 1]  = cy - 0.5f * h;
    boxes[g*4 + 2]  = cx + 0.5f * w;
    boxes[g*4 + 3]  = cy + 0.5f * h;
    cls[g]          = bc;
  }
}

// ---------------------------------------------------------------------------
// Kernel 2: exact stable top-1000 per image via full 32K bitonic sort in LDS.
// Packed key: ordered(float)<<32 | (0xFFFFFFFF - idx)  -> descending sort is
// (score desc, idx asc), matching lax.top_k tie-breaking.
// 256 KB LDS per workgroup -- fits in CDNA5's 320 KB WGP LDS.
// ---------------------------------------------------------------------------
__global__ __launch_bounds__(1024) void k_topk(const float* __restrict__ score,
                                               float* __restrict__ tk_score,
                                               int*   __restrict__ tk_idx) {
  extern __shared__ unsigned long long keys[];        // SORT_N entries
  const int img = blockIdx.x;
  const float* s = score + (size_t)img * NPRED;

  for (int i = threadIdx.x; i < SORT_N; i += blockDim.x) {
    float v = (i < NPRED) ? s[i] : -__builtin_huge_valf();
    unsigned u = __float_as_uint(v);
    u = (u & 0x80000000u) ? ~u : (u | 0x80000000u);   // total order
    keys[i] = ((unsigned long long)u << 32) | (unsigned)(0xFFFFFFFFu - (unsigned)i);
  }
  __syncthreads();

  for (int k = 2; k <= SORT_N; k <<= 1) {
    for (int j = k >> 1; j > 0; j >>= 1) {
      for (int i = threadIdx.x; i < SORT_N; i += blockDim.x) {
        int l = i ^ j;
        if (l > i) {
          bool desc = ((i & k) == 0);
          unsigned long long a = keys[i], b = keys[l];
          if ((a < b) == desc) { keys[i] = b; keys[l] = a; }
        }
      }
      __syncthreads();
    }
  }

  for (int i = threadIdx.x; i < KTOP; i += blockDim.x) {
    unsigned long long p = keys[i];
    unsigned u = (unsigned)(p >> 32);
    u = (u & 0x80000000u) ? (u & 0x7FFFFFFFu) : ~u;   // undo ordering map
    tk_score[(size_t)img * KTOP + i] = __uint_as_float(u);
    tk_idx  [(size_t)img * KTOP + i] = (int)(0xFFFFFFFFu - (unsigned)(p & 0xFFFFFFFFu));
  }
}

// ---------------------------------------------------------------------------
// Kernel 3: bitmask NMS.
//  Phase B (parallel, 1024 thr): full 1000x1000 suppression bit-matrix in LDS
//           (125 KB -- only possible with CDNA5's 320 KB LDS).
//  Phase C (single wave32):     greedy reduction, lane l owns supp word l,
//           v_readlane broadcasts the decision bit -- no barriers in loop.
//  Output: 32-word keep bitmask per image.
// ---------------------------------------------------------------------------
__global__ __launch_bounds__(1024) void k_nms(const float* __restrict__ boxes,
                                              const float* __restrict__ tk_score,
                                              const int*   __restrict__ tk_idx,
                                              unsigned*    __restrict__ keepBits) {
  extern __shared__ unsigned char smemRaw[];
  float*    X1 = (float*)smemRaw;
  float*    Y1 = X1 + KTOP;
  float*    X2 = Y1 + KTOP;
  float*    Y2 = X2 + KTOP;
  float*    AR = Y2 + KTOP;
  unsigned* maskRow = (unsigned*)(AR + KTOP);         // KTOP * NWORD words
  __shared__ unsigned suppInit[NWORD];

  const int img = blockIdx.x;
  const int t   = threadIdx.x;

  if (t < NWORD) suppInit[t] = 0u;
  __syncthreads();

  if (t < KTOP) {
    const int   idx = tk_idx[(size_t)img * KTOP + t];
    const float sc  = tk_score[(size_t)img * KTOP + t];
    const float* bb = boxes + ((size_t)img * NPRED + idx) * 4;
    const float x1 = bb[0], y1 = bb[1], x2 = bb[2], y2 = bb[3];
    X1[t] = x1; Y1[t] = y1; X2[t] = x2; Y2[t] = y2;
    AR[t] = (x2 - x1) * (y2 - y1);
    if (!(sc > 0.0f)) atomicOr(&suppInit[t >> 5], 1u << (t & 31));
  }
  __syncthreads();

  // Phase B: one 32-bit word of the suppression matrix per loop step.
  for (int w = t; w < KTOP * NWORD; w += blockDim.x) {
    const int i  = w >> 5;
    const int j0 = (w & 31) << 5;
    const float x1 = X1[i], y1 = Y1[i], x2 = X2[i], y2 = Y2[i], ai = AR[i];
    unsigned bits = 0u;
    #pragma unroll 4
    for (int b = 0; b < 32; ++b) {
      const int j = j0 + b;
      if (j < KTOP && j > i) {
        float lx = fmaxf(x1, X1[j]);
        float ly = fmaxf(y1, Y1[j]);
        float rx = fminf(x2, X2[j]);
        float ry = fminf(y2, Y2[j]);
        float inter = fmaxf(rx - lx, 0.0f) * fmaxf(ry - ly, 0.0f);
        float iou   = inter / (ai + AR[j] - inter + 1e-9f);
        if (iou > IOU_T) bits |= 1u << b;
      }
    }
    maskRow[w] = bits;
  }
  __syncthreads();

  // Phase C: wave 0 only, lane t owns columns [32t, 32t+31].
  if (t < 32) {
    unsigned supp  = suppInit[t];
    unsigned keepw = 0u;
    for (int i = 0; i < KTOP; ++i) {
      const unsigned wi = __builtin_amdgcn_readlane(supp, i >> 5);
      const bool alive  = ((wi >> (i & 31)) & 1u) == 0u;
      const unsigned row = maskRow[i * NWORD + t];
      if (alive) {
        supp |= row;
        if (t == (i >> 5)) keepw |= 1u << (i & 31);
      }
    }
    keepBits[img * NWORD + t] = keepw;
  }
}

// ---------------------------------------------------------------------------
// Kernel 4: stable compaction (wave32 popcount scan over keep bitmask) ->
// first min(#kept,300) rows, zero-fill, det/mask writes, and logits gather
// via the Tensor Data Mover in GATHER mode (16 rows of 320B per descriptor,
// 16-bit row indices, tracked on TENSORcnt), staged through LDS.
// ---------------------------------------------------------------------------
__global__ __launch_bounds__(256) void k_final(const float* __restrict__ boxes,
                                               const float* __restrict__ tk_score,
                                               const int*   __restrict__ tk_idx,
                                               const int*   __restrict__ cls,
                                               const unsigned* __restrict__ keepBits,
                                               const float* __restrict__ logits,
                                               float*       __restrict__ out) {
  extern __shared__ float gbuf[];                     // MAXDET*NCLS floats
  __shared__ int sel[MAXDET];
  __shared__ int selCnt;
  const int img = blockIdx.x;
  const int t   = threadIdx.x;

  float* det  = out + (size_t)img * MAXDET * 6;
  float* mout = out + (size_t)NB * MAXDET * 6 + (size_t)img * MAXDET;
  float* lout = out + (size_t)NB * MAXDET * 6 + (size_t)NB * MAXDET
                    + (size_t)img * MAXDET * NCLS;

  for (int i = t; i < MAXDET * 6;    i += blockDim.x) det[i]  = 0.0f;
  for (int i = t; i < MAXDET;        i += blockDim.x) mout[i] = 0.0f;
  for (int i = t; i < MAXDET * NCLS; i += blockDim.x) lout[i] = 0.0f;

  // Stable compaction: wave32 scan over 32 keep words.
  if (t < 32) {
    int base = 0;
    for (int c = 0; c < NWORD; ++c) {
      const unsigned wrd = keepBits[img * NWORD + c];
      const bool f = (wrd >> t) & 1u;
      const int pos = base + __popc(wrd & ((1u << t) - 1u));
      const int i = c * 32 + t;
      if (f && pos < MAXDET) sel[pos] = i;
      base += __popc(wrd);
    }
    if (t == 0) selCnt = (base < MAXDET) ? base : MAXDET;
  }
  __syncthreads();
  const int cnt = selCnt;

  // TDM gather: one lane builds descriptors; DMA engine fetches rows -> LDS.
  if (t == 0) {
    const unsigned gbufBase = (unsigned)(uintptr_t)gbuf;
    const unsigned long long gaddr =
        (unsigned long long)(uintptr_t)(logits + (size_t)img * NPRED * NCLS);
    const int ngrp = (cnt + 15) >> 4;
    for (int g = 0; g < ngrp; ++g) {
      const int n = min(16, cnt - g * 16);
      unsigned idxw[8];
      for (int w = 0; w < 8; ++w) {
        unsigned lo = 0, hi = 0;
        const int k0 = 2 * w, k1 = 2 * w + 1;
        if (k0 < n) lo = (unsigned)tk_idx[(size_t)img * KTOP + sel[g * 16 + k0]] & 0xFFFFu;
        if (k1 < n) hi = (unsigned)tk_idx[(size_t)img * KTOP + sel[g * 16 + k1]] & 0xFFFFu;
        idxw[w] = lo | (hi << 16);
      }
      // D# group 0: count=1, gather_mode=1 (16-bit idx), lds_addr, global_addr, type=2
      u32x4 g0;
      g0.x = 1u | (1u << 31);
      g0.y = gbufBase + (unsigned)g * 16u * NCLS * 4u;
      g0.z = (unsigned)(gaddr & 0xFFFFFFFFull);
      g0.w = (unsigned)((gaddr >> 32) & 0x1FFFFFFull) | (2u << 30);
      // D# group 1: data_size=4B, tensor_dim0=80, tensor_dim1=NPRED,
      //             tile_dim0=80, tile_dim1=#indices, dim0_stride=80
      i32x8 g1;
      g1[0] = (int)(2u << 16);          // data_size = 2 -> 4 bytes
      g1[1] = (int)((unsigned)NCLS << 16);    // tensor_dim0[15:0]
      g1[2] = (int)((unsigned)NPRED << 16);   // tensor_dim0[31:16]=0 | tensor_dim1[15:0]
      g1[3] = (int)((unsigned)NCLS << 16);    // tensor_dim1[31:16]=0 | tile_dim0
      g1[4] = n;                               // tile_dim1 = #valid indices
      g1[5] = NCLS;                            // tensor_dim0_stride (elements)
      g1[6] = 0;
      g1[7] = 0;
      // D# groups 2/3: sixteen 16-bit row indices
      i32x4 g2, g3;
      g2[0] = (int)idxw[0]; g2[1] = (int)idxw[1]; g2[2] = (int)idxw[2]; g2[3] = (int)idxw[3];
      g3[0] = (int)idxw[4]; g3[1] = (int)idxw[5]; g3[2] = (int)idxw[6]; g3[3] = (int)idxw[7];
      // Unused trailing group (VADDR4 slot) -- zero.
      i32x8 g4;
      g4[0] = 0; g4[1] = 0; g4[2] = 0; g4[3] = 0;
      g4[4] = 0; g4[5] = 0; g4[6] = 0; g4[7] = 0;
      __builtin_amdgcn_tensor_load_to_lds(g0, g1, g2, g3, g4, 0);
    }
    __builtin_amdgcn_s_wait_tensorcnt(0);
  }
  __syncthreads();

  for (int r = t; r < cnt; r += blockDim.x) {
    const int p   = sel[r];
    const int idx = tk_idx[(size_t)img * KTOP + p];
    const float* bb = boxes + ((size_t)img * NPRED + idx) * 4;
    det[r*6 + 0] = bb[0];
    det[r*6 + 1] = bb[1];
    det[r*6 + 2] = bb[2];
    det[r*6 + 3] = bb[3];
    det[r*6 + 4] = tk_score[(size_t)img * KTOP + p];
    det[r*6 + 5] = (float)cls[(size_t)img * NPRED + idx];
    mout[r] = 1.0f;
  }
  __syncthreads();
  for (int w = t; w < cnt * NCLS; w += blockDim.x)
    lout[w] = gbuf[w];
}

// ---------------------------------------------------------------------------
extern "C" void kernel_launch(void* const* d_in, const int* in_sizes, int n_in,
                              void* d_out, int out_size, void* d_ws, size_t ws_size,
                              hipStream_t stream) {
  const float* pred   = (const float*)d_in[0];  // (16,25200,85)
  const float* logits = (const float*)d_in[1];  // (16,25200,80)
  float* out = (float*)d_out;

  char* ws = (char*)d_ws;
  size_t off = 0;
  float*    score   = (float*)(ws + off);    off += (size_t)NB * NPRED * 4;
  float*    boxesW  = (float*)(ws + off);    off += (size_t)NB * NPRED * 4 * 4;
  int*      clsW    = (int*)(ws + off);      off += (size_t)NB * NPRED * 4;
  float*    tks     = (float*)(ws + off);    off += (size_t)NB * KTOP * 4;
  int*      tki     = (int*)(ws + off);      off += (size_t)NB * KTOP * 4;
  unsigned* keepB   = (unsigned*)(ws + off); off += (size_t)NB * NWORD * 4;
  (void)in_sizes; (void)n_in; (void)out_size; (void)ws_size;

  dim3 gPrep((NPRED + TILE_ROWS - 1) / TILE_ROWS, NB);
  const size_t ldsPrep = (size_t)TILE_ROWS * ROWF * sizeof(float);       //  87 KB
  const size_t ldsSort = (size_t)SORT_N * sizeof(unsigned long long);    // 256 KB
  const size_t ldsNms  = (size_t)(5 * KTOP) * sizeof(float)
                       + (size_t)(KTOP * NWORD) * sizeof(unsigned);      // 148 KB
  const size_t ldsFin  = (size_t)MAXDET * NCLS * sizeof(float);          //  96 KB

  k_prep <<<gPrep, TILE_ROWS, ldsPrep, stream>>>(pred, score, boxesW, clsW);
  k_topk <<<NB, 1024, ldsSort, stream>>>(score, tks, tki);
  k_nms  <<<NB, 1024, ldsNms,  stream>>>(boxesW, tks, tki, keepB);
  k_final<<<NB, 256,  ldsFin,  stream>>>(boxesW, tks, tki, clsW, keepB, logits, out);
}